// Smoothness2DLoss_4037269258979
// MI455X (gfx1250) — compile-verified
//
#include <hip/hip_runtime.h>
#include <stdint.h>

#define BS 32
#define NT 50
#define NP 3
#define TP 150
#define NH 4
#define NK 88                 // 44 keypoints * 2 coords
#define PTS_PER_B (TP * NK)   // 13200 floats per batch
#define VEC4_PER_B (PTS_PER_B / 4)
#define KCHUNKS 22            // 88 / 4 float4 chunks per row

typedef float v2f __attribute__((ext_vector_type(2)));
typedef float v4f __attribute__((ext_vector_type(4)));
typedef float v8f __attribute__((ext_vector_type(8)));

// Wave32 sum-reduction via V_WMMA_F32_16X16X4_F32 with B = ones.
// A: VGPR0 = acc (lanes 0-15 -> (M=lane,K=0), lanes 16-31 -> (M=lane-16,K=2)),
// VGPR1 = 0. D[m,n] = acc[m] + acc[m+16] for every n. Summing the 8 D
// components per lane then one shfl_xor(16) gives the 32-lane sum in all lanes.
__device__ __forceinline__ float wave_sum32(float v) {
  v2f a; a.x = v;   a.y = 0.0f;
  v2f b; b.x = 1.0f; b.y = 1.0f;
  v8f c = {};
  c = __builtin_amdgcn_wmma_f32_16x16x4_f32(
      /*neg_a=*/false, a, /*neg_b=*/false, b,
      /*c_mod=*/(short)0, c, /*reuse_a=*/false, /*reuse_b=*/false);
  float s = c[0] + c[1] + c[2] + c[3] + c[4] + c[5] + c[6] + c[7];
  s += __shfl_xor(s, 16, 32);
  return s;
}

// Branch-free per-lane partial of the L1 distance between row i (preloaded in
// pi) and row j: lanes 0..21 each own one aligned float4 chunk of the 88
// coords; lanes 22..31 read chunk 0 and are zeroed by kmask.
__device__ __forceinline__ float l1_partial(const float* __restrict__ s_pts,
                                            int j, int koff, float kmask,
                                            v4f pi) {
  const v4f pj = *(const v4f*)(&s_pts[j * NK + koff]);
  const float d = fabsf(pi.x - pj.x) + fabsf(pi.y - pj.y) +
                  fabsf(pi.z - pj.z) + fabsf(pi.w - pj.w);
  return d * kmask;
}

// One block (256 threads = 8 waves) per batch. Stages points_2d[b] into LDS
// with async-to-LDS DMA, then each wave processes rows i = wave, wave+8, ...
// Band mask (closed form): live j have |frame(j)-frame(i)|<=1 and
// max(frame) <= 48; frame-49 rows use the uniform 1/150 softmax.
__global__ void __launch_bounds__(256) smooth_num_kernel(
    const int* __restrict__ idx, const float* __restrict__ pts,
    const float* __restrict__ attn, float* __restrict__ ws_num) {
  __shared__ float s_pts[PTS_PER_B];
  __shared__ float s_idxf[TP];
  __shared__ float s_red[8];

  const int b    = blockIdx.x;
  const int tid  = threadIdx.x;
  const int lane = tid & 31;
  const int wave = tid >> 5;

  // --- async DMA of points_2d[b] (52.8 KB) into LDS -------------------------
  {
    const uint64_t gbase = (uint64_t)(uintptr_t)(pts + (size_t)b * PTS_PER_B);
    const uint32_t lbase = (uint32_t)(uintptr_t)(&s_pts[0]);
    for (int u = tid; u < VEC4_PER_B; u += 256) {
      uint32_t loff = lbase + (uint32_t)u * 16u;
      uint32_t goff = (uint32_t)u * 16u;
      asm volatile("global_load_async_to_lds_b128 %0, %1, %2"
                   :: "v"(loff), "v"(goff), "s"(gbase) : "memory");
    }
  }
  if (tid < TP) s_idxf[tid] = (float)idx[b * TP + tid];
  asm volatile("s_wait_asynccnt 0" ::: "memory");
  __syncthreads();

  float acc = 0.0f;  // per-lane partial of the numerator
  const int   koff  = (lane < KCHUNKS) ? 4 * lane : 0;  // aligned float4 chunk
  const float kmask = (lane < KCHUNKS) ? 1.0f : 0.0f;

  for (int i = wave; i < TP; i += 8) {
    const float idx_i = s_idxf[i];           // wave-uniform
    if (idx_i == 0.0f) continue;
    const int ti = i / 3;
    const v4f pi = *(const v4f*)(&s_pts[i * NK + koff]);

    if (ti <= 48) {
      const int f0 = (ti > 0)  ? ti - 1 : 0;
      const int f1 = (ti < 48) ? ti + 1 : 48;
      const int j0 = 3 * f0;
      const int nb = 3 * (f1 - f0 + 1);      // <= 9 live columns

      float w[9];
      #pragma unroll
      for (int jj = 0; jj < 9; ++jj) w[jj] = 0.0f;

      // band softmax per head (exact: masked exps underflow to 0), sum heads
      for (int h = 0; h < NH; ++h) {
        const float* arow =
            attn + ((((size_t)b * NH + h) * TP + i) * TP + j0);
        float av[9];
        float m = -3.0e38f;
        for (int jj = 0; jj < nb; ++jj) { av[jj] = arow[jj]; m = fmaxf(m, av[jj]); }
        float S = 0.0f;
        for (int jj = 0; jj < nb; ++jj) { av[jj] = __expf(av[jj] - m); S += av[jj]; }
        const float invS = 1.0f / S;
        for (int jj = 0; jj < nb; ++jj) w[jj] += av[jj] * invS;
      }

      for (int jj = 0; jj < nb; ++jj) {
        const int j = j0 + jj;
        const float c = w[jj] * s_idxf[j] * idx_i;   // wave-uniform
        acc += l1_partial(s_pts, j, koff, kmask, pi) * c;
      }
    } else {
      // frame 49: fully masked row -> uniform softmax 1/150 over all heads
      const float c0 = idx_i * ((float)NH / (float)TP);
      for (int j = 0; j < TP; ++j) {
        const float cj = c0 * s_idxf[j];             // wave-uniform
        acc += l1_partial(s_pts, j, koff, kmask, pi) * cj;
      }
    }
  }

  // wave reduction via WMMA, then fixed-order cross-wave combine
  const float wsum = wave_sum32(acc);
  if (lane == 0) s_red[wave] = wsum;
  __syncthreads();
  if (tid == 0) {
    float s = 0.0f;
    for (int w2 = 0; w2 < 8; ++w2) s += s_red[w2];
    ws_num[b] = s;
  }
}

// Single block: deterministic denominator + WMMA reduction of the 32 batch
// partials, then the final scalar divide.
__global__ void __launch_bounds__(256) smooth_final_kernel(
    const int* __restrict__ idx, const float* __restrict__ ws_num,
    float* __restrict__ out) {
  __shared__ float s_den[8];
  const int tid  = threadIdx.x;
  const int lane = tid & 31;
  const int wave = tid >> 5;

  // den = sum_b (sum_i idx[b,i])^2
  float dacc = 0.0f;
  for (int b = wave; b < BS; b += 8) {
    int s = 0;
    for (int c = lane; c < TP; c += 32) s += idx[b * TP + c];
    #pragma unroll
    for (int m = 16; m >= 1; m >>= 1) s += __shfl_xor(s, m, 32);
    if (lane == 0) dacc += (float)s * (float)s;
  }
  if (lane == 0) s_den[wave] = dacc;
  __syncthreads();

  // num = sum of 32 per-batch partials (all waves run the wmma; wave 0 real)
  const float v = (tid < BS) ? ws_num[tid] : 0.0f;
  const float num = wave_sum32(v);

  if (tid == 0) {
    float den = 0.0f;
    for (int w2 = 0; w2 < 8; ++w2) den += s_den[w2];
    out[0] = num / (1.0f + (float)NH * den);
  }
}

extern "C" void kernel_launch(void* const* d_in, const int* in_sizes, int n_in,
                              void* d_out, int out_size, void* d_ws, size_t ws_size,
                              hipStream_t stream) {
  (void)in_sizes; (void)n_in; (void)out_size; (void)ws_size;
  const int*   idx  = (const int*)d_in[0];   // (32,150) int32
  // d_in[1] idx_mask and d_in[2] idx_kp are unused by the reference
  const float* pts  = (const float*)d_in[3]; // (32,150,44,2) f32
  const float* attn = (const float*)d_in[4]; // (32,4,150,150) f32
  float* out = (float*)d_out;
  float* ws  = (float*)d_ws;                 // 32 floats of per-batch partials

  smooth_num_kernel<<<BS, 256, 0, stream>>>(idx, pts, attn, ws);
  smooth_final_kernel<<<1, 256, 0, stream>>>(idx, ws, out);
}